// OuterProductNetworkLayer_59476707115127
// MI455X (gfx1250) — compile-verified
//
#include <hip/hip_runtime.h>

typedef float v2f __attribute__((ext_vector_type(2)));
typedef float v8f __attribute__((ext_vector_type(8)));

constexpr int kBatch  = 2048;
constexpr int kFields = 32;
constexpr int kEmbed  = 64;
constexpr int kPairs  = 496;   // 32*31/2
constexpr int kBTile  = 128;   // batch rows per workgroup (8 waves x 16)
constexpr int kStride = 68;    // padded LDS row stride (floats): banks 4*row+col -> conflict-free b64 reads

__global__ __launch_bounds__(256, 2)
void opnl_wmma_kernel(const float* __restrict__ emb,
                      const float* __restrict__ kern,
                      float* __restrict__ out)
{
    __shared__ float sK[kEmbed * kStride];   // K_p stored [i][j], padded rows
    __shared__ float sP[kBTile * kStride];   // P tile stored [m][j], padded rows

    const int tid  = threadIdx.x;
    const int wave = tid >> 5;
    const int lane = tid & 31;
    const int lm   = lane & 15;
    const int hi   = lane >> 4;     // wave half: K-offset 0 or 2 inside each 4-step
    const int koff = 2 * hi;

    const int p  = blockIdx.x >> 4;            // pair index 0..495
    const int b0 = (blockIdx.x & 15) * kBTile; // batch tile origin

    // pair index -> (row_field, col_field) of triu(32, k=1)
    int pr = p, rf = 0;
    while (pr >= (kFields - 1 - rf)) { pr -= (kFields - 1 - rf); ++rf; }
    const int cf = rf + 1 + pr;

    // ---- stage K_p (64x64, kern[i, p, j]) into LDS, coalesced float4 ----
    {
        const float* src = kern + (size_t)p * kEmbed;
        #pragma unroll
        for (int e = 0; e < 4; ++e) {
            int idx4 = tid + e * 256;          // 0..1023 float4s
            int i    = idx4 >> 4;
            int j4   = (idx4 & 15) << 2;
            float4 v = *(const float4*)(src + (size_t)i * (kPairs * kEmbed) + j4);
            *(float4*)(&sK[i * kStride + j4]) = v;
        }
    }
    // ---- stage P tile (128x64, emb[b0+m, rf, j]) into LDS, coalesced float4 ----
    {
        #pragma unroll
        for (int e = 0; e < 8; ++e) {
            int idx4 = tid + e * 256;          // 0..2047 float4s
            int m    = idx4 >> 4;
            int j4   = (idx4 & 15) << 2;
            float4 v = *(const float4*)(emb + (size_t)(b0 + m) * (kFields * kEmbed)
                                        + rf * kEmbed + j4);
            *(float4*)(&sP[m * kStride + j4]) = v;
        }
    }
    __syncthreads();

    // ---- per-wave: KP(16 x 64) = P(16x64) x K_p^T via V_WMMA_F32_16X16X4_F32 ----
    // A (16x4): lanes 0-15 hold M=lm, K={k0,k0+1}; lanes 16-31 hold M=lm, K={k0+2,k0+3}
    // B (4x16): lanes 0-15 hold N=lm, K={k0,k0+1}; lanes 16-31 hold N=lm, K={k0+2,k0+3}
    //           B[k][n] = K_p[i=n][j=k]  -> read sK row n at column k
    const int mbase = wave * 16;
    v8f acc0 = {}, acc1 = {}, acc2 = {}, acc3 = {};

    const float* aPtr  = sP + (mbase + lm) * kStride + koff;
    const float* bPtr0 = sK + (lm +  0) * kStride + koff;
    const float* bPtr1 = sK + (lm + 16) * kStride + koff;
    const float* bPtr2 = sK + (lm + 32) * kStride + koff;
    const float* bPtr3 = sK + (lm + 48) * kStride + koff;

    #pragma unroll
    for (int ks = 0; ks < 16; ++ks) {
        const int k0 = ks * 4;
        v2f a  = *(const v2f*)(aPtr  + k0);
        v2f b0v = *(const v2f*)(bPtr0 + k0);
        v2f b1v = *(const v2f*)(bPtr1 + k0);
        v2f b2v = *(const v2f*)(bPtr2 + k0);
        v2f b3v = *(const v2f*)(bPtr3 + k0);
        acc0 = __builtin_amdgcn_wmma_f32_16x16x4_f32(false, a, false, b0v, (short)0, acc0, false, false);
        acc1 = __builtin_amdgcn_wmma_f32_16x16x4_f32(false, a, false, b1v, (short)0, acc1, false, false);
        acc2 = __builtin_amdgcn_wmma_f32_16x16x4_f32(false, a, false, b2v, (short)0, acc2, false, false);
        acc3 = __builtin_amdgcn_wmma_f32_16x16x4_f32(false, a, false, b3v, (short)0, acc3, false, false);
    }

    // ---- epilogue: out[b, p] = sum_i KP[b, i] * Q[b, i] ----
    // D layout: VGPR r of lane holds (M = r + 8*hi, N = lm); N-tile t adds 16*t to i.
    const int mrow = mbase + 8 * hi;
    float part[8];
    #pragma unroll
    for (int r = 0; r < 8; ++r) {
        const int m = mrow + r;
        const float* qRow = emb + (size_t)(b0 + m) * (kFields * kEmbed) + cf * kEmbed;
        float s = acc0[r] * qRow[ 0 + lm]
                + acc1[r] * qRow[16 + lm]
                + acc2[r] * qRow[32 + lm]
                + acc3[r] * qRow[48 + lm];
        // reduce over the 16 lanes (i within each N-tile) of this wave half
        s += __shfl_xor(s, 1, 16);
        s += __shfl_xor(s, 2, 16);
        s += __shfl_xor(s, 4, 16);
        s += __shfl_xor(s, 8, 16);
        part[r] = s;
    }
    if (lm == 0) {
        #pragma unroll
        for (int r = 0; r < 8; ++r) {
            out[(size_t)(b0 + mrow + r) * kPairs + p] = part[r];
        }
    }
}

extern "C" void kernel_launch(void* const* d_in, const int* in_sizes, int n_in,
                              void* d_out, int out_size, void* d_ws, size_t ws_size,
                              hipStream_t stream) {
    (void)in_sizes; (void)n_in; (void)d_ws; (void)ws_size; (void)out_size;
    const float* emb  = (const float*)d_in[0];   // (2048, 32, 64) fp32
    const float* kern = (const float*)d_in[1];   // (64, 496, 64) fp32
    float* out = (float*)d_out;                  // (2048, 1, 496) fp32

    dim3 grid(kPairs * (kBatch / kBTile));       // 496 * 16 = 7936 blocks
    opnl_wmma_kernel<<<grid, dim3(256), 0, stream>>>(emb, kern, out);
}